// GCNConv_69002944577596
// MI455X (gfx1250) — compile-verified
//
#include <hip/hip_runtime.h>
#include <hip/hip_bf16.h>

#define N_NODES 100000
#define N_EDGES 1600000
#define D_IN    128
#define D_OUT   128

typedef __attribute__((ext_vector_type(2))) float    v2f;
typedef __attribute__((ext_vector_type(8))) float    v8f;
typedef __attribute__((ext_vector_type(4))) unsigned v4u;
typedef __attribute__((ext_vector_type(8))) unsigned v8u;

// ---------------------------------------------------------------------------
// TDM: DMA a 2D fp32 tile (tile_d0 x tile_d1, row stride stride0 elements)
// from global memory into LDS at byte offset lds_off. D# layout per CDNA5 ISA
// ch.8: group0 = {count/mode, lds_addr, global_addr[56:0], type=2},
// group1 = {data_size=4B, tensor_dim0/1, tile_dim0/1/2, tensor_dim0_stride}.
// Issued with inline asm (toolchain-portable); tracked by TENSORcnt.
// ---------------------------------------------------------------------------
__device__ __forceinline__ void tdm_load_2d_f32(unsigned lds_off, const void* gptr,
                                                unsigned tensor_d0, unsigned tensor_d1,
                                                unsigned tile_d0, unsigned tile_d1,
                                                unsigned stride0_elems) {
    unsigned long long ga = (unsigned long long)(uintptr_t)gptr;
    v4u g0;
    g0[0] = 1u;                                            // count=1, user D#
    g0[1] = lds_off;                                       // LDS byte address
    g0[2] = (unsigned)(ga & 0xFFFFFFFFu);                  // global_addr[31:0]
    g0[3] = (unsigned)((ga >> 32) & 0x01FFFFFFu)           // global_addr[56:32]
          | (2u << 30);                                    // type = 2 ("image")
    v8u g1;
    g1[0] = 2u << 16;                                      // data_size = 4 bytes
    g1[1] = (tensor_d0 & 0xFFFFu) << 16;                   // tensor_dim0[15:0]
    g1[2] = (tensor_d0 >> 16) | ((tensor_d1 & 0xFFFFu) << 16); // d0[31:16], d1[15:0]
    g1[3] = (tensor_d1 >> 16) | (tile_d0 << 16);           // d1[31:16], tile_dim0
    g1[4] = tile_d1 & 0xFFFFu;                             // tile_dim1 (tile_dim2=0)
    g1[5] = stride0_elems;                                 // tensor_dim0_stride[31:0]
    g1[6] = 0u;
    g1[7] = 0u;
    asm volatile("tensor_load_to_lds %0, %1"
                 :: "s"(g0), "s"(g1)
                 : "memory");
}

// ---------------------------------------------------------------------------
// Kernel 1: out[n, :] = bias[:]
// ---------------------------------------------------------------------------
__global__ void gcn_bias_init(const float* __restrict__ bias,
                              float* __restrict__ out) {
    int i = blockIdx.x * blockDim.x + threadIdx.x;          // float4 index
    float4 b4 = ((const float4*)bias)[i & 31];              // 128 cols = 32 float4
    ((float4*)out)[i] = b4;
}

// ---------------------------------------------------------------------------
// Kernel 2: src_h = x @ W1 via V_WMMA_F32_16X16X4_F32, operands staged in LDS
// by the Tensor Data Mover. Block = 256 thr = 8 waves; block owns 16 rows,
// wave w owns 16-col tile w. 32 WMMA k-steps of K=4 cover D_IN=128.
// ---------------------------------------------------------------------------
__global__ void gcn_gemm_f32(const float* __restrict__ x,
                             const float* __restrict__ w,
                             float* __restrict__ src_h) {
    __shared__ float sW[D_IN * D_OUT];   // 64 KB: full weight matrix
    __shared__ float sA[16 * D_IN];      // 8 KB: this block's 16-row x tile

    const int lane = threadIdx.x & 31;
    const int wave = threadIdx.x >> 5;
    const int row0 = blockIdx.x << 4;
    const int col0 = wave << 4;
    const int half = lane >> 4;          // 0: lanes 0-15, 1: lanes 16-31
    const int l16  = lane & 15;

    if (wave == 0) {                     // one DMA issue per block (per-wave op)
        tdm_load_2d_f32((unsigned)(uintptr_t)sW, w,
                        D_OUT, D_IN, D_OUT, D_IN, D_OUT);
        tdm_load_2d_f32((unsigned)(uintptr_t)sA, x + (size_t)row0 * D_IN,
                        D_IN, 16, D_IN, 16, D_IN);
        __builtin_amdgcn_s_wait_tensorcnt(0);
    }
    __syncthreads();                     // publish LDS to all 8 waves

    v8f acc = {};
    #pragma unroll
    for (int k0 = 0; k0 < D_IN; k0 += 4) {
        const int ka = k0 + (half << 1);
        // A 16x4: lane l16 = row, two consecutive K per lane (ds_load_b64)
        v2f a = *(const v2f*)(sA + l16 * D_IN + ka);
        // B 4x16: rows ka, ka+1 striped across N = col0 + l16
        v2f b;
        b.x = sW[ka * D_OUT + col0 + l16];
        b.y = sW[(ka + 1) * D_OUT + col0 + l16];
        acc = __builtin_amdgcn_wmma_f32_16x16x4_f32(
            false, a, false, b, (short)0, acc, false, false);
    }

    // D 16x16 f32: VGPR r -> (M = r + 8*half, N = l16)
    #pragma unroll
    for (int r = 0; r < 8; ++r) {
        src_h[(size_t)(row0 + r + (half << 3)) * D_OUT + col0 + l16] = acc[r];
    }
}

// ---------------------------------------------------------------------------
// Kernel 3: one wave per edge; float4 gather of src_h[edge_row], scale,
// native f32 global atomic scatter-add into out[edge_col] (L2-resident).
// ---------------------------------------------------------------------------
__global__ void gcn_scatter(const float* __restrict__ src_h,
                            const float* __restrict__ edge_vals,
                            const int*   __restrict__ edge_row,
                            const int*   __restrict__ edge_col,
                            float* __restrict__ out) {
    const unsigned gid  = blockIdx.x * blockDim.x + threadIdx.x;
    const unsigned e    = gid >> 5;          // edge index (uniform per wave)
    const unsigned lane = gid & 31;
    if (e >= N_EDGES) return;

    const int   r = edge_row[e];
    const int   c = edge_col[e];
    const float v = edge_vals[e];

    const float4 s = ((const float4*)(src_h + (size_t)r * D_OUT))[lane];
    float* dst = out + (size_t)c * D_OUT + (lane << 2);
    unsafeAtomicAdd(dst + 0, s.x * v);
    unsafeAtomicAdd(dst + 1, s.y * v);
    unsafeAtomicAdd(dst + 2, s.z * v);
    unsafeAtomicAdd(dst + 3, s.w * v);
}

// ---------------------------------------------------------------------------
extern "C" void kernel_launch(void* const* d_in, const int* in_sizes, int n_in,
                              void* d_out, int out_size, void* d_ws, size_t ws_size,
                              hipStream_t stream) {
    const float* x         = (const float*)d_in[0];
    const float* edge_vals = (const float*)d_in[1];
    const float* weight1   = (const float*)d_in[2];
    const float* bias1     = (const float*)d_in[3];
    const int*   edge_row  = (const int*)d_in[4];
    const int*   edge_col  = (const int*)d_in[5];
    float*       out       = (float*)d_out;
    float*       src_h     = (float*)d_ws;   // 100000*128*4 = 51.2 MB scratch

    // 1) out = broadcast(bias)
    {
        int n4 = N_NODES * D_OUT / 4;                     // 3,200,000 float4
        gcn_bias_init<<<n4 / 256, 256, 0, stream>>>(bias1, out);
    }
    // 2) src_h = x @ W1 (WMMA f32, TDM-staged operands)
    {
        int blocks = N_NODES / 16;                        // 6250 (exact)
        gcn_gemm_f32<<<blocks, 256, 0, stream>>>(x, weight1, src_h);
    }
    // 3) gather/scale/scatter-add over edges
    {
        long long threads = (long long)N_EDGES * 32;      // wave per edge
        int blocks = (int)(threads / 256);                // 200,000
        gcn_scatter<<<blocks, 256, 0, stream>>>(src_h, edge_vals,
                                                edge_row, edge_col, out);
    }
}